// TransformerLayer_60705067762276
// MI455X (gfx1250) — compile-verified
//
#include <hip/hip_runtime.h>
#include <math.h>

// Nystromformer layer for MI455X (gfx1250), fp32 via V_WMMA_F32_16X16X4_F32.
#define DIMV 512
#define NSEQ 4096
#define HEADS 8
#define DHEAD 64
#define LMK 16    // landmarks l = n/m
#define MWIN 256  // window m
#define QSCALE 0.125f
#define LNEPS 1e-5f
#define T1CH 16   // split-K chunks for attn3 @ v

typedef __attribute__((ext_vector_type(2))) float v2f;
typedef __attribute__((ext_vector_type(8))) float v8f;

__device__ __forceinline__ v8f vzero8() {
  v8f r;
#pragma unroll
  for (int i = 0; i < 8; ++i) r[i] = 0.0f;
  return r;
}

__device__ __forceinline__ v8f wmma4(v2f a, v2f b, v8f c) {
  return __builtin_amdgcn_wmma_f32_16x16x4_f32(false, a, false, b, (short)0, c,
                                               false, false);
}

// Accumulate C(16x16) += A(16xK)*B(Kx16).
// A row-major, K contiguous: elem(m,k) = A[m*lda + k]
// B generic strides:         elem(k,n) = B[k*ldbr + n*ldbc]
// Fragment layout (ISA 7.12.2, wave32):
//   A 16x4 f32: lane m = lane&15; VGPR0/1 hold K = 2*(lane>>4) + {0,1}
//   B 4x16 f32: lane n = lane&15; VGPR0/1 hold K = 2*(lane>>4) + {0,1}
//   C 16x16:    VGPR r -> M = r + 8*(lane>>4), N = lane&15
__device__ __forceinline__ v8f wmma_loop(const float* __restrict__ A, int lda,
                                         const float* __restrict__ B, int ldbr,
                                         int ldbc, int K, v8f c) {
  const int lane = threadIdx.x & 31;
  const int lo = lane & 15;
  const int hi = (lane >> 4) << 1;
  const float* ap = A + (size_t)lo * lda + hi;
  const float* bp = B + (size_t)hi * ldbr + (size_t)lo * ldbc;
#pragma unroll 4
  for (int k0 = 0; k0 < K; k0 += 4) {
    v2f a;
    a.x = ap[0];
    a.y = ap[1];
    v2f bb;
    bb.x = bp[0];
    bb.y = bp[ldbr];
    c = wmma4(a, bb, c);
    ap += 4;
    bp += (size_t)4 * ldbr;
  }
  return c;
}

// 64x16 macro-tile: 4 row-tiles share one B fragment per K-step (A from global).
template <int MT>
__device__ __forceinline__ void wmma_loop_mt(const float* __restrict__ A,
                                             int lda,
                                             const float* __restrict__ B,
                                             int ldbr, int ldbc, int K,
                                             v8f* acc) {
  const int lane = threadIdx.x & 31;
  const int lo = lane & 15;
  const int hi = (lane >> 4) << 1;
  const float* ap = A + (size_t)lo * lda + hi;
  const float* bp = B + (size_t)hi * ldbr + (size_t)lo * ldbc;
  for (int k0 = 0; k0 < K; k0 += 4) {
    v2f bb;
    bb.x = bp[0];
    bb.y = bp[ldbr];
#pragma unroll
    for (int t = 0; t < MT; ++t) {
      v2f a;
      a.x = ap[(size_t)t * 16 * lda];
      a.y = ap[(size_t)t * 16 * lda + 1];
      acc[t] = wmma4(a, bb, acc[t]);
    }
    ap += 4;
    bp += (size_t)4 * ldbr;
  }
}

// Block-cooperative GEMM: 8 waves share one 64-row A panel, staged through LDS
// in 64x32 chunks (stride 36 floats: 16B-aligned rows, conflict-free frags).
// Each wave accumulates its own 64x16 C tile against a private B column tile.
#define ASTRIDE 36
__device__ __forceinline__ void gemm_block_a_staged(
    const float* __restrict__ A, int lda, int K, const float* __restrict__ B,
    int ldbr, int ldbc, v8f* acc) {
  __shared__ float As[64 * ASTRIDE];  // 9216 B
  const int tid = threadIdx.x;
  const int lane = tid & 31;
  const int lo = lane & 15;
  const int hi = (lane >> 4) << 1;
  const float* bp = B + (size_t)hi * ldbr + (size_t)lo * ldbc;
  const int r = tid >> 2;         // 0..63: row this thread copies
  const int cc = (tid & 3) << 3;  // 0,8,16,24: first of 8 cols it copies
  for (int kc = 0; kc < K; kc += 32) {
    __syncthreads();  // previous chunk fully consumed
    {
      const float4* src = (const float4*)(A + (size_t)r * lda + kc + cc);
      float4* dst = (float4*)(&As[r * ASTRIDE + cc]);
      dst[0] = src[0];
      dst[1] = src[1];
    }
    __syncthreads();  // chunk visible to all waves
#pragma unroll
    for (int kk = 0; kk < 32; kk += 4) {
      v2f bb;
      bb.x = bp[0];
      bb.y = bp[ldbr];
#pragma unroll
      for (int t = 0; t < 4; ++t) {
        v2f a;
        a.x = As[(t * 16 + lo) * ASTRIDE + kk + hi];
        a.y = As[(t * 16 + lo) * ASTRIDE + kk + hi + 1];
        acc[t] = wmma4(a, bb, acc[t]);
      }
      bp += (size_t)4 * ldbr;
    }
  }
}

__device__ __forceinline__ void store_tile(float* C, int ldr, int ldc, v8f c) {
  const int lane = threadIdx.x & 31;
  const int n = lane & 15;
  const int m0 = (lane >> 4) << 3;
#pragma unroll
  for (int r = 0; r < 8; ++r)
    C[(size_t)(m0 + r) * ldr + (size_t)n * ldc] = c[r];
}

// dst[m][n] = alpha*I[m][n] - c   (16x16 row-major, e.g. LDS)
__device__ __forceinline__ void store_aIm(float* dst, float alpha, v8f c) {
  const int lane = threadIdx.x & 31;
  const int n = lane & 15;
  const int m0 = (lane >> 4) << 3;
#pragma unroll
  for (int r = 0; r < 8; ++r) {
    const int m = m0 + r;
    dst[m * 16 + n] = ((m == n) ? alpha : 0.0f) - c[r];
  }
}

// ---------------- LayerNorm: one block (256 thr) per 512-wide row ----------
__global__ void ln_kernel(const float* __restrict__ x, const float* __restrict__ g,
                          const float* __restrict__ be, float* __restrict__ nrm) {
  const int row = blockIdx.x;
  const int tid = threadIdx.x;
  const float* xr = x + (size_t)row * DIMV;
  const float a = xr[tid];
  const float b2 = xr[tid + 256];
  __shared__ float red[256];
  red[tid] = a + b2;
  __syncthreads();
  for (int s = 128; s > 0; s >>= 1) {
    if (tid < s) red[tid] += red[tid + s];
    __syncthreads();
  }
  const float mu = red[0] * (1.0f / DIMV);
  __syncthreads();
  const float da = a - mu, db = b2 - mu;
  red[tid] = da * da + db * db;
  __syncthreads();
  for (int s = 128; s > 0; s >>= 1) {
    if (tid < s) red[tid] += red[tid + s];
    __syncthreads();
  }
  const float rstd = rsqrtf(red[0] * (1.0f / DIMV) + LNEPS);
  float* o = nrm + (size_t)row * DIMV;
  o[tid] = da * rstd * g[tid] + be[tid];
  o[tid + 256] = db * rstd * g[tid + 256] + be[tid + 256];
}

// ------------- QKV GEMM: (nrow x 512)@(512 x 1536), scatter to b,h,n,d -----
__global__ void qkv_kernel(const float* __restrict__ nrm,
                           const float* __restrict__ w, float* __restrict__ q,
                           float* __restrict__ k, float* __restrict__ v) {
  const int nt = blockIdx.x * 8 + (threadIdx.x >> 5);  // 0..95 col tile
  const int mg = blockIdx.y;                           // 64-row group
  v8f acc[4] = {vzero8(), vzero8(), vzero8(), vzero8()};
  gemm_block_a_staged(nrm + (size_t)mg * 64 * DIMV, DIMV, DIMV, w + nt * 16,
                      3 * DIMV, 1, acc);
  const int c0 = nt * 16;
  const int which = c0 >> 9;  // 0=q 1=k 2=v
  const int h = (c0 & 511) >> 6;
  const int d0 = c0 & 63;
  float* dst = (which == 0) ? q : (which == 1) ? k : v;
  const float sc = (which == 0) ? QSCALE : 1.0f;
  const int gr0 = mg * 64;
  const int bb = gr0 >> 12;
  const int nn0 = gr0 & (NSEQ - 1);
  const int bh = bb * HEADS + h;
#pragma unroll
  for (int t = 0; t < 4; ++t) {
    v8f cc = acc[t];
#pragma unroll
    for (int i = 0; i < 8; ++i) cc[i] *= sc;
    float* Ct = dst + ((size_t)bh * NSEQ + nn0 + t * 16) * DHEAD + d0;
    store_tile(Ct, DHEAD, 1, cc);
  }
}

// ------------- landmark means over windows of 256 --------------------------
__global__ void landmark_kernel(const float* __restrict__ q,
                                const float* __restrict__ k,
                                float* __restrict__ ql, float* __restrict__ kl) {
  const int idx = blockIdx.x * blockDim.x + threadIdx.x;  // (bh, j, d)
  const int d = idx & 63;
  const int j = (idx >> 6) & 15;
  const int bh = idx >> 10;
  const size_t base = ((size_t)bh * NSEQ + j * MWIN) * DHEAD + d;
  float sq = 0.f, sk = 0.f;
  for (int i = 0; i < MWIN; ++i) {
    sq += q[base + (size_t)i * DHEAD];
    sk += k[base + (size_t)i * DHEAD];
  }
  ql[idx] = sq * (1.0f / MWIN);
  kl[idx] = sk * (1.0f / MWIN);
}

// ------------- sim1 = q @ kl^T : per bh (4096x64)@(64x16) ------------------
__global__ void sim1_kernel(const float* __restrict__ q,
                            const float* __restrict__ kl,
                            float* __restrict__ s1) {
  const int bh = blockIdx.y;
  const int mg = blockIdx.x * 8 + (threadIdx.x >> 5);  // 0..63
  v8f acc[4] = {vzero8(), vzero8(), vzero8(), vzero8()};
  const float* A = q + ((size_t)bh * NSEQ + mg * 64) * DHEAD;
  const float* B = kl + (size_t)bh * LMK * DHEAD;  // elem(d,j)=kl[j*64+d]
  wmma_loop_mt<4>(A, DHEAD, B, 1, DHEAD, DHEAD, acc);
  float* C = s1 + ((size_t)bh * NSEQ + mg * 64) * LMK;
#pragma unroll
  for (int t = 0; t < 4; ++t) store_tile(C + t * 16 * LMK, LMK, 1, acc[t]);
}

// ------------- sim2 = ql @ kl^T : one wave per bh, 16x16 -------------------
__global__ void sim2_kernel(const float* __restrict__ ql,
                            const float* __restrict__ kl,
                            float* __restrict__ s2) {
  const int bh = blockIdx.x * 8 + (threadIdx.x >> 5);
  v8f acc = wmma_loop(ql + (size_t)bh * LMK * DHEAD, DHEAD,
                      kl + (size_t)bh * LMK * DHEAD, 1, DHEAD, DHEAD, vzero8());
  store_tile(s2 + (size_t)bh * 256, 16, 1, acc);
}

// ------------- sim3 = ql @ k^T : per bh (16x64)@(64x4096) ------------------
__global__ void sim3_kernel(const float* __restrict__ ql,
                            const float* __restrict__ k,
                            float* __restrict__ s3) {
  const int bh = blockIdx.y;
  const int nt = blockIdx.x * 8 + (threadIdx.x >> 5);  // 0..255
  const float* A = ql + (size_t)bh * LMK * DHEAD;
  const float* B = k + ((size_t)bh * NSEQ + nt * 16) * DHEAD;  // elem(d,j)=k[j*64+d]
  v8f acc = wmma_loop(A, DHEAD, B, 1, DHEAD, DHEAD, vzero8());
  store_tile(s3 + (size_t)bh * LMK * NSEQ + nt * 16, NSEQ, 1, acc);
}

// ------------- softmax over rows of 16 (one thread per row) ----------------
__global__ void softmax16_kernel(float* __restrict__ p, int nrows) {
  const int r = blockIdx.x * blockDim.x + threadIdx.x;
  if (r >= nrows) return;
  float* row = p + (size_t)r * 16;
  float mx = row[0];
#pragma unroll
  for (int j = 1; j < 16; ++j) mx = fmaxf(mx, row[j]);
  float e[16];
  float s = 0.f;
#pragma unroll
  for (int j = 0; j < 16; ++j) {
    e[j] = __expf(row[j] - mx);
    s += e[j];
  }
  const float inv = 1.0f / s;
#pragma unroll
  for (int j = 0; j < 16; ++j) row[j] = e[j] * inv;
}

// ------------- softmax over rows of 4096 (one block per row) ---------------
__global__ void softmax_long_kernel(float* __restrict__ p) {
  float* row = p + (size_t)blockIdx.x * NSEQ;
  const int tid = threadIdx.x;
  __shared__ float red[256];
  float mx = -3.4e38f;
  for (int i = tid; i < NSEQ; i += 256) mx = fmaxf(mx, row[i]);
  red[tid] = mx;
  __syncthreads();
  for (int s = 128; s > 0; s >>= 1) {
    if (tid < s) red[tid] = fmaxf(red[tid], red[tid + s]);
    __syncthreads();
  }
  mx = red[0];
  __syncthreads();
  float sum = 0.f;
  for (int i = tid; i < NSEQ; i += 256) {
    const float e = __expf(row[i] - mx);
    row[i] = e;
    sum += e;
  }
  red[tid] = sum;
  __syncthreads();
  for (int s = 128; s > 0; s >>= 1) {
    if (tid < s) red[tid] += red[tid + s];
    __syncthreads();
  }
  const float inv = 1.0f / red[0];
  for (int i = tid; i < NSEQ; i += 256) row[i] *= inv;
}

// ------------- Moore-Penrose pinv of 16x16, one wave per bh ----------------
__global__ void pinv_kernel(const float* __restrict__ a2,
                            float* __restrict__ zout) {
  __shared__ float xs[256], zs[256], ts[256], us[256];
  __shared__ float rsums[16], csums[16], dsh;
  const int bh = blockIdx.x;
  const int lane = threadIdx.x;  // block of exactly 32
  const float* src = a2 + (size_t)bh * 256;
  for (int i = lane; i < 256; i += 32) xs[i] = src[i];
  __syncthreads();
  if (lane < 16) {
    float rsum = 0.f, csum = 0.f;
    for (int j = 0; j < 16; ++j) {
      rsum += fabsf(xs[lane * 16 + j]);
      csum += fabsf(xs[j * 16 + lane]);
    }
    rsums[lane] = rsum;
    csums[lane] = csum;
  }
  __syncthreads();
  if (lane == 0) {
    float mc = 0.f, mr = 0.f;
    for (int j = 0; j < 16; ++j) {
      mc = fmaxf(mc, rsums[j]);
      mr = fmaxf(mr, csums[j]);
    }
    float dd = mc * mr;
    if (dd == 0.f) dd = 1e-8f;
    dsh = 1.0f / dd;
  }
  __syncthreads();
  const float dn = dsh;
  for (int i = lane; i < 256; i += 32) {
    const int m = i >> 4, n = i & 15;
    zs[i] = xs[n * 16 + m] * dn;  // z = x^T / (max_col*max_row)
  }
  __syncthreads();
  for (int it = 0; it < 6; ++it) {
    v8f t = wmma_loop(xs, 16, zs, 16, 1, 16, vzero8());  // t = x@z
    store_tile(ts, 16, 1, t);
    __syncthreads();
    for (int i = lane; i < 256; i += 32) {
      const int m = i >> 4, n = i & 15;
      us[i] = ((m == n) ? 7.0f : 0.0f) - ts[i];  // 7I - t
    }
    __syncthreads();
    v8f w = wmma_loop(ts, 16, us, 16, 1, 16, vzero8());  // t@(7I-t)
    __syncthreads();
    store_aIm(us, 15.0f, w);  // 15I - ...
    __syncthreads();
    v8f w2 = wmma_loop(ts, 16, us, 16, 1, 16, vzero8());
    __syncthreads();
    store_aIm(us, 13.0f, w2);  // 13I - ...
    __syncthreads();
    v8f zn = wmma_loop(zs, 16, us, 16, 1, 16, vzero8());  // z@(13I-...)
    __syncthreads();
    {
      const int n = lane & 15;
      const int m0 = (lane >> 4) << 3;
#pragma unroll
      for (int r = 0; r < 8; ++r) zs[(m0 + r) * 16 + n] = 0.25f * zn[r];
    }
    __syncthreads();
  }
  float* dst = zout + (size_t)bh * 256;
  for (int i = lane; i < 256; i += 32) dst[i] = zs[i];
}

// ------------- t1 = attn3 @ v, split-K over 16 chunks of 256 ---------------
__global__ void t1_split_kernel(const float* __restrict__ a3,
                                const float* __restrict__ v,
                                float* __restrict__ part) {
  const int bh = blockIdx.x;
  const int ch = blockIdx.y;        // 0..T1CH-1
  const int nt = threadIdx.x >> 5;  // 0..3 (block of 128)
  const int k0 = ch * (NSEQ / T1CH);
  const float* A = a3 + (size_t)bh * LMK * NSEQ + k0;
  const float* B = v + ((size_t)bh * NSEQ + k0) * DHEAD + nt * 16;
  v8f acc = wmma_loop(A, NSEQ, B, DHEAD, 1, NSEQ / T1CH, vzero8());
  store_tile(part + ((size_t)bh * T1CH + ch) * (LMK * DHEAD) + nt * 16, DHEAD,
             1, acc);
}

__global__ void t1_reduce_kernel(const float* __restrict__ part,
                                 float* __restrict__ t1) {
  const int idx = blockIdx.x * blockDim.x + threadIdx.x;  // bh*1024 total
  const int bh = idx >> 10;
  const int i = idx & 1023;
  float s = 0.f;
#pragma unroll
  for (int c = 0; c < T1CH; ++c)
    s += part[((size_t)bh * T1CH + c) * (LMK * DHEAD) + i];
  t1[idx] = s;
}

// ------------- t2 = zinv @ t1 : per bh (16x16)@(16x64) ---------------------
__global__ void t2_kernel(const float* __restrict__ z,
                          const float* __restrict__ t1,
                          float* __restrict__ t2) {
  const int bh = blockIdx.x;
  const int nt = threadIdx.x >> 5;
  v8f acc = wmma_loop(z + (size_t)bh * 256, 16,
                      t1 + (size_t)bh * LMK * DHEAD + nt * 16, DHEAD, 1, 16,
                      vzero8());
  store_tile(t2 + (size_t)bh * LMK * DHEAD + nt * 16, DHEAD, 1, acc);
}

// ------------- out = attn1 @ t2 + v, scattered to (b, n, 512) --------------
__global__ void outattn_kernel(const float* __restrict__ a1,
                               const float* __restrict__ t2,
                               const float* __restrict__ v,
                               float* __restrict__ obnd) {
  const int bh = blockIdx.y;
  const int id = blockIdx.x * 8 + (threadIdx.x >> 5);  // 0..255
  const int mg = id >> 2;                              // 64-row group
  const int nt = id & 3;                               // 16-col tile
  v8f acc[4] = {vzero8(), vzero8(), vzero8(), vzero8()};
  const float* A = a1 + ((size_t)bh * NSEQ + mg * 64) * LMK;
  const float* B = t2 + (size_t)bh * LMK * DHEAD + nt * 16;
  wmma_loop_mt<4>(A, LMK, B, DHEAD, 1, LMK, acc);
  const int bb = bh >> 3, h = bh & 7;
  const int lane = threadIdx.x & 31;
  const int n = lane & 15;
  const int m0 = (lane >> 4) << 3;
#pragma unroll
  for (int t = 0; t < 4; ++t) {
#pragma unroll
    for (int r = 0; r < 8; ++r) {
      const int nn = mg * 64 + t * 16 + m0 + r;
      const int d = nt * 16 + n;
      const float val = acc[t][r] + v[((size_t)bh * NSEQ + nn) * DHEAD + d];
      obnd[((size_t)bb * NSEQ + nn) * DIMV + h * DHEAD + d] = val;
    }
  }
}

// ------------- final: out = x + obnd @ w_out + b_out -----------------------
__global__ void final_kernel(const float* __restrict__ obnd,
                             const float* __restrict__ wout,
                             const float* __restrict__ bout,
                             const float* __restrict__ x,
                             float* __restrict__ out) {
  const int nt = blockIdx.x * 8 + (threadIdx.x >> 5);  // 0..31
  const int mg = blockIdx.y;
  v8f acc[4] = {vzero8(), vzero8(), vzero8(), vzero8()};
  gemm_block_a_staged(obnd + (size_t)mg * 64 * DIMV, DIMV, DIMV, wout + nt * 16,
                      DIMV, 1, acc);
  const int lane = threadIdx.x & 31;
  const int n = lane & 15;
  const int m0 = (lane >> 4) << 3;
  const int c = nt * 16 + n;
  const float bb = bout[c];
#pragma unroll
  for (int t = 0; t < 4; ++t) {
#pragma unroll
    for (int r = 0; r < 8; ++r) {
      const size_t gr = (size_t)mg * 64 + t * 16 + m0 + r;
      out[gr * DIMV + c] = x[gr * DIMV + c] + acc[t][r] + bb;
    }
  }
}

extern "C" void kernel_launch(void* const* d_in, const int* in_sizes, int n_in,
                              void* d_out, int out_size, void* d_ws,
                              size_t ws_size, hipStream_t stream) {
  const float* x = (const float*)d_in[0];
  const float* g = (const float*)d_in[1];
  const float* be = (const float*)d_in[2];
  const float* wqkv = (const float*)d_in[3];
  const float* wout = (const float*)d_in[4];
  const float* bout = (const float*)d_in[5];
  float* out = (float*)d_out;

  const int b = in_sizes[0] / (NSEQ * DIMV);
  const int nrow = b * NSEQ;
  const int bh = b * HEADS;

  float* ws = (float*)d_ws;
  size_t off = 0;
  float* nrm = ws + off;  off += (size_t)nrow * DIMV;
  float* q = ws + off;    off += (size_t)nrow * DIMV;
  float* k = ws + off;    off += (size_t)nrow * DIMV;
  float* v = ws + off;    off += (size_t)nrow * DIMV;
  float* s1 = ws + off;   off += (size_t)bh * NSEQ * LMK;
  float* s3 = ws + off;   off += (size_t)bh * LMK * NSEQ;
  float* s2 = ws + off;   off += (size_t)bh * 256;
  float* zm = ws + off;   off += (size_t)bh * 256;
  float* ql = ws + off;   off += (size_t)bh * LMK * DHEAD;
  float* kl = ws + off;   off += (size_t)bh * LMK * DHEAD;
  float* t1 = ws + off;   off += (size_t)bh * LMK * DHEAD;
  float* t2 = ws + off;   off += (size_t)bh * LMK * DHEAD;
  float* part = ws + off; off += (size_t)bh * T1CH * LMK * DHEAD;
  float* obnd = nrm;  // nrm is dead after the QKV GEMM; reuse as attn output

  ln_kernel<<<nrow, 256, 0, stream>>>(x, g, be, nrm);
  qkv_kernel<<<dim3(12, nrow / 64), 256, 0, stream>>>(nrm, wqkv, q, k, v);
  landmark_kernel<<<(bh * LMK * DHEAD) / 256, 256, 0, stream>>>(q, k, ql, kl);
  sim1_kernel<<<dim3(8, bh), 256, 0, stream>>>(q, kl, s1);
  sim2_kernel<<<bh / 8, 256, 0, stream>>>(ql, kl, s2);
  sim3_kernel<<<dim3(32, bh), 256, 0, stream>>>(ql, k, s3);
  softmax16_kernel<<<(bh * NSEQ + 255) / 256, 256, 0, stream>>>(s1, bh * NSEQ);
  softmax16_kernel<<<(bh * LMK + 255) / 256, 256, 0, stream>>>(s2, bh * LMK);
  softmax_long_kernel<<<bh * LMK, 256, 0, stream>>>(s3);
  pinv_kernel<<<bh, 32, 0, stream>>>(s2, zm);
  t1_split_kernel<<<dim3(bh, T1CH), 128, 0, stream>>>(s3, v, part);
  t1_reduce_kernel<<<(bh * LMK * DHEAD) / 256, 256, 0, stream>>>(part, t1);
  t2_kernel<<<bh, 128, 0, stream>>>(zm, t1, t2);
  outattn_kernel<<<dim3(32, bh), 256, 0, stream>>>(s1, t2, v, obnd);
  final_kernel<<<dim3(4, nrow / 64), 256, 0, stream>>>(obnd, wout, bout, x, out);
}